// PositionalEmbedding_60155311947993
// MI455X (gfx1250) — compile-verified
//
#include <hip/hip_runtime.h>
#include <cstdint>

// Problem constants (match reference).
#define B_ROWS   131072
#define D_OUT    512
#define NCOMP    6
// log2(2100) / 256  (phase exponent scale): inv_freq[f] = 2^(-(2f+1)*K2)
#define K2       0.04311005831f
#define INV_2PI  0.15915494309189535f

// Launch geometry: wave32; 8 waves/block, 8 rows/wave -> 64 rows/block.
#define WAVES            8
#define ROWS_PER_WAVE    8
#define ROWS_PER_BLOCK   (WAVES * ROWS_PER_WAVE)   // 64
#define THREADS          (WAVES * 32)              // 256

// Clang-native 4-float vector (usable with __builtin_nontemporal_store).
typedef float v4f __attribute__((ext_vector_type(4)));

__global__ __launch_bounds__(THREADS)
void pe_sincos_kernel(const int* __restrict__ comp,     // [B, 6] int32
                      const float* __restrict__ params, // [6]    f32
                      float* __restrict__ out)          // [B, 512] f32
{
    __shared__ int smem[ROWS_PER_BLOCK * NCOMP];        // 1536 B

    const int tid      = threadIdx.x;
    const int lane     = tid & 31;
    const int wv       = tid >> 5;
    const int waveRow  = blockIdx.x * ROWS_PER_BLOCK + wv * ROWS_PER_WAVE;

    // ---------------------------------------------------------------
    // CDNA5 async global->LDS stage: this wave's 8 rows x 6 ints = 192 B,
    // moved as 12 x b128 transfers (lanes 0..11). Tracked by ASYNCcnt.
    // Each wave stages only its own rows -> a per-wave s_wait_asynccnt
    // suffices, no workgroup barrier required.
    // ---------------------------------------------------------------
    {
        // Low 32 bits of a flat shared pointer == LDS byte offset on gfx1250.
        unsigned ldsBase = (unsigned)(uintptr_t)(&smem[wv * ROWS_PER_WAVE * NCOMP]);
        if (lane < 12) {
            unsigned    ldsOff = ldsBase + (unsigned)(lane * 16);
            const char* gaddr  = (const char*)(comp + (size_t)waveRow * NCOMP) + lane * 16;
            asm volatile("global_load_async_to_lds_b128 %0, %1, off"
                         :: "v"(ldsOff), "v"(gaddr)
                         : "memory");
        }
        asm volatile("s_wait_asynccnt 0" ::: "memory");
    }

    // Per-component learned scalars (uniform -> scalar loads).
    float p[NCOMP];
#pragma unroll
    for (int k = 0; k < NCOMP; ++k) p[k] = params[k];

    // Per-lane inverse frequencies with 1/(2*pi) folded in, so the raw
    // v_sin_f32 / v_cos_f32 instructions (which compute sin/cos(2*pi*x))
    // need no per-call range-scale multiply. Amortized over all 8 rows.
    // Pass i, lane l covers f0 = 64*i + 2*l and f1 = f0 + 1.
    float invf0[4], invf1[4];
#pragma unroll
    for (int i = 0; i < 4; ++i) {
        const int f0 = 64 * i + 2 * lane;
        invf0[i] = INV_2PI * exp2f(-K2 * (float)(2 * f0 + 1));   // v_exp_f32
        invf1[i] = INV_2PI * exp2f(-K2 * (float)(2 * f0 + 3));
    }

    const float inv6 = 1.0f / 6.0f;

#pragma unroll
    for (int r = 0; r < ROWS_PER_WAVE; ++r) {
        const int  row = waveRow + r;
        const int* c   = &smem[(wv * ROWS_PER_WAVE + r) * NCOMP];

        // scaled[k] = params[k] * comp[row][k]  (LDS broadcast reads)
        float sc[NCOMP];
#pragma unroll
        for (int k = 0; k < NCOMP; ++k) sc[k] = p[k] * (float)c[k];

        float* orow = out + (size_t)row * D_OUT;

#pragma unroll
        for (int i = 0; i < 4; ++i) {
            float s0 = 0.f, c0 = 0.f, s1 = 0.f, c1 = 0.f;
#pragma unroll
            for (int k = 0; k < NCOMP; ++k) {
                // angle in "revolutions": a = sc[k]*inv_freq[f] / (2*pi)
                const float a0 = sc[k] * invf0[i];
                const float a1 = sc[k] * invf1[i];
                s0 += __builtin_amdgcn_sinf(a0);   // v_sin_f32: sin(2*pi*a0)
                c0 += __builtin_amdgcn_cosf(a0);   // v_cos_f32
                s1 += __builtin_amdgcn_sinf(a1);
                c1 += __builtin_amdgcn_cosf(a1);
            }
            // out[row, 2f] = mean sin, out[row, 2f+1] = mean cos, f = f0, f0+1
            // -> one contiguous 16B vector at column 128*i + 4*lane.
            v4f v = { s0 * inv6, c0 * inv6, s1 * inv6, c1 * inv6 };
            // Write-once 256MB stream > 192MB L2: non-temporal b128 store.
            __builtin_nontemporal_store(v, (v4f*)(orow + i * 128 + lane * 4));
        }
    }
}

extern "C" void kernel_launch(void* const* d_in, const int* in_sizes, int n_in,
                              void* d_out, int out_size, void* d_ws, size_t ws_size,
                              hipStream_t stream)
{
    (void)in_sizes; (void)n_in; (void)d_ws; (void)ws_size; (void)out_size;

    const int*   comp   = (const int*)d_in[0];    // [B, 6] int32
    const float* params = (const float*)d_in[1];  // [6, 1] f32
    float*       out    = (float*)d_out;          // [B, 512] f32

    const int grid = B_ROWS / ROWS_PER_BLOCK;     // 2048 blocks
    pe_sincos_kernel<<<grid, THREADS, 0, stream>>>(comp, params, out);
}